// TriPlaneEncoder_47966194762056
// MI455X (gfx1250) — compile-verified
//
#include <hip/hip_runtime.h>
#include <cmath>

#define R_PLANE 4096
#define NLEVELS 8

typedef float v4f __attribute__((ext_vector_type(4)));

struct ScaleParams { float s[NLEVELS]; };

// One thread per (position b, level l).  Lane layout: t = b*8 + l, so the 8
// lanes covering one position write one contiguous 256B output row; a wave32
// writes a fully coalesced 1KB tile.  12 float2 gathers per thread, issued
// back-to-back so a single s_wait_loadcnt covers them.
__global__ __launch_bounds__(256) void triplane_fwd(
    const float* __restrict__ pos,      // [B,3]
    const float* __restrict__ planes,   // [3, R*R, 2] flat
    float* __restrict__ out,            // [B, 64]
    ScaleParams sp, int B)
{
    int t = blockIdx.x * 256 + threadIdx.x;
    int b = t >> 3;
    if (b >= B) return;
    int l = t & 7;

    float p3[3];
    p3[0] = pos[3 * b + 0];
    p3[1] = pos[3 * b + 1];
    p3[2] = pos[3 * b + 2];

    const float s   = sp.s[l];
    const float sm1 = s - 1.0f;

    // Per-axis grid/frac and the two clipped big-grid coordinates (b=0,1).
    float fr[3];
    int   gi[3][2];
#pragma unroll
    for (int j = 0; j < 3; ++j) {
        float p = fmaf(p3[j], sm1, 0.5f);
        float g = floorf(p);
        fr[j] = p - g;
#pragma unroll
        for (int bb = 0; bb < 2; ++bb) {
            float q = (g + (float)bb) / s * (float)R_PLANE;  // match ref: /s then *R, f32
            int gq = (int)q;                                  // trunc, values >= 0
            gq = gq < 0 ? 0 : (gq > (R_PLANE - 1) ? (R_PLANE - 1) : gq);
            gi[j][bb] = gq;
        }
    }

    // plane p uses axes (A0[p], A1[p]):  d0=[1,0,0], d1=[2,2,1]
    const int A0[3] = {1, 0, 0};
    const int A1[3] = {2, 2, 1};

    const float2* __restrict__ pl = (const float2*)planes;   // [3*R*R] float2

    // Issue all 12 corner gathers (global_load_b64) before consuming any.
    float2 c[3][4];
#pragma unroll
    for (int p = 0; p < 3; ++p) {
        const int base = p * (R_PLANE * R_PLANE);
        const int x0 = gi[A0[p]][0], x1 = gi[A0[p]][1];
        const int y0 = gi[A1[p]][0], y1 = gi[A1[p]][1];
        c[p][0] = pl[(size_t)(base + x0 + y0 * R_PLANE)];   // (b0=0,b1=0)
        c[p][1] = pl[(size_t)(base + x0 + y1 * R_PLANE)];   // (b0=0,b1=1)
        c[p][2] = pl[(size_t)(base + x1 + y0 * R_PLANE)];   // (b0=1,b1=0)
        c[p][3] = pl[(size_t)(base + x1 + y1 * R_PLANE)];   // (b0=1,b1=1)
    }

    float fx[3], fy[3];
#pragma unroll
    for (int p = 0; p < 3; ++p) {
        const float f0 = fr[A0[p]], f1 = fr[A1[p]];
        const float w0 = (1.0f - f0) * (1.0f - f1);
        const float w1 = (1.0f - f0) * f1;
        const float w2 = f0 * (1.0f - f1);
        const float w3 = f0 * f1;
        float ax = w0 * c[p][0].x;
        float ay = w0 * c[p][0].y;
        ax = fmaf(w1, c[p][1].x, ax);  ay = fmaf(w1, c[p][1].y, ay);
        ax = fmaf(w2, c[p][2].x, ax);  ay = fmaf(w2, c[p][2].y, ay);
        ax = fmaf(w3, c[p][3].x, ax);  ay = fmaf(w3, c[p][3].y, ay);
        fx[p] = ax; fy[p] = ay;
    }

    const float prx = fx[0] * fx[1] * fx[2];
    const float pry = fy[0] * fy[1] * fy[2];

    // out row for (b,l): [f0x f0y f1x f1y f2x f2y prodx prody]
    v4f o0 = { fx[0], fy[0], fx[1], fy[1] };
    v4f o1 = { fx[2], fy[2], prx, pry };
    v4f* op = (v4f*)(out + (size_t)b * 64 + (size_t)l * 8);
    // Streaming 256MB output: non-temporal stores (TH=NT global_store_b128)
    // keep L2 free for the small-level gather working set (~70MB resident in
    // the 192MB L2).
    __builtin_nontemporal_store(o0, op);
    __builtin_nontemporal_store(o1, op + 1);
}

extern "C" void kernel_launch(void* const* d_in, const int* in_sizes, int n_in,
                              void* d_out, int out_size, void* d_ws, size_t ws_size,
                              hipStream_t stream) {
    const float* positions = (const float*)d_in[0];
    const float* planes    = (const float*)d_in[1];
    float*       out       = (float*)d_out;

    const int B = in_sizes[0] / 3;

    // Replicate numpy's _plane_scales in double precision on the host.
    ScaleParams sp;
    const double lb = log((double)R_PLANE / 16.0) / (double)(NLEVELS - 1);
    const double bb = exp(lb);
    const double lnb = log(bb);
    for (int i = 0; i < NLEVELS; ++i) {
        double v = ceil(16.0 * exp((double)i * lnb) - 1.0) + 1.0;
        sp.s[i] = (float)v;
    }

    const long long total  = (long long)B * NLEVELS;
    const int       blocks = (int)((total + 255) / 256);
    hipLaunchKernelGGL(triplane_fwd, dim3(blocks), dim3(256), 0, stream,
                       positions, planes, out, sp, B);
}